// GNN3_1614907703642
// MI455X (gfx1250) — compile-verified
//
#include <hip/hip_runtime.h>

typedef __attribute__((ext_vector_type(2))) float v2f;
typedef __attribute__((ext_vector_type(8))) float v8f;

#define N_NODES 100000
#define F_DIM   128
#define E_EDGES 1600000

// ---------------------------------------------------------------- zero
__global__ void gnn_zero_kernel(float* __restrict__ p, int n) {
    int i = blockIdx.x * blockDim.x + threadIdx.x;
    if (i < n) p[i] = 0.0f;
}

// ---------------------------------------------------------------- degree
__global__ void gnn_degree_kernel(const int* __restrict__ dst,
                                  float* __restrict__ deg, int ne) {
    int e = blockIdx.x * blockDim.x + threadIdx.x;
    if (e < ne) unsafeAtomicAdd(&deg[dst[e]], 1.0f);
}

// ------------------------------------------------- scatter-add x[src] -> agg[dst]
// one thread = (edge, 4 contiguous features); 32 threads cover one edge's 128 feats
__global__ void gnn_scatter_kernel(const float* __restrict__ feat,
                                   const int* __restrict__ src,
                                   const int* __restrict__ dst,
                                   float* __restrict__ agg, int ne) {
    int gid = blockIdx.x * blockDim.x + threadIdx.x;
    int e = gid >> 5;
    if (e >= ne) return;
    int c4 = (gid & 31) << 2;                       // feature offset 0..124
    const float4 v = *(const float4*)(feat + (size_t)src[e] * F_DIM + c4);
    float* dp = agg + (size_t)dst[e] * F_DIM + c4;
    unsafeAtomicAdd(dp + 0, v.x);
    unsafeAtomicAdd(dp + 1, v.y);
    unsafeAtomicAdd(dp + 2, v.z);
    unsafeAtomicAdd(dp + 3, v.w);
}

// ------------------------------------------------- fused SAGE layer (f32 WMMA)
// out = relu?( (agg/deg) @ Wl + xin @ Wr + bias )
// one wave computes a 16-row x 128-col output stripe (8 x 16x16 D tiles).
__global__ __launch_bounds__(256) void gnn_sage_gemm_kernel(
    const float* __restrict__ agg,   // [N,128] summed neighbor feats
    const float* __restrict__ deg,   // [N]
    const float* __restrict__ xin,   // [N,128] root feats
    const float* __restrict__ Wl,    // [128,128] row-major
    const float* __restrict__ Wr,    // [128,128]
    const float* __restrict__ bias,  // [128]
    float* __restrict__ out,         // [N,128] (may alias xin; row-wise safe)
    int relu, int nrows)
{
    int wid  = (blockIdx.x * blockDim.x + threadIdx.x) >> 5;
    int lane = threadIdx.x & 31;
    int row0 = wid * 16;
    if (row0 >= nrows) return;                       // wave-uniform exit

    int laneM = lane & 15;                           // M (and N) index within tile
    int koff  = (lane >> 4) << 1;                    // 0 for lanes 0-15, 2 for 16-31
    int row   = row0 + laneM;
    float dscale = 1.0f / fmaxf(deg[row], 1.0f);     // mean = agg * dscale

    v8f acc[8];
#pragma unroll
    for (int j = 0; j < 8; ++j) acc[j] = (v8f){};

    const float* arow_m = agg + (size_t)row * F_DIM + koff;
    const float* arow_x = xin + (size_t)row * F_DIM + koff;

    for (int k = 0; k < F_DIM; k += 4) {
        // A fragments (16x4 f32): lane<16 holds K=k..k+1, lane>=16 K=k+2..k+3
        v2f am = { arow_m[k] * dscale, arow_m[k + 1] * dscale };
        v2f ax = { arow_x[k],          arow_x[k + 1] };
#pragma unroll
        for (int j = 0; j < 8; ++j) {
            int col = (j << 4) + laneM;
            // B fragments (4x16 f32): row K=k+koff / k+koff+1, N=col
            v2f bl = { Wl[(k + koff) * F_DIM + col], Wl[(k + koff + 1) * F_DIM + col] };
            acc[j] = __builtin_amdgcn_wmma_f32_16x16x4_f32(
                false, am, false, bl, (short)0, acc[j], false, false);
            v2f br = { Wr[(k + koff) * F_DIM + col], Wr[(k + koff + 1) * F_DIM + col] };
            acc[j] = __builtin_amdgcn_wmma_f32_16x16x4_f32(
                false, ax, false, br, (short)0, acc[j], false, false);
        }
    }

    // D layout: VGPR r -> row row0 + r + (lane<16 ? 0 : 8), col = j*16 + (lane&15)
    int mhalf = (lane >> 4) << 3;
#pragma unroll
    for (int j = 0; j < 8; ++j) {
        int col = (j << 4) + laneM;
        float bb = bias[col];
#pragma unroll
        for (int r = 0; r < 8; ++r) {
            float v = acc[j][r] + bb;
            if (relu) v = fmaxf(v, 0.0f);
            out[(size_t)(row0 + mhalf + r) * F_DIM + col] = v;
        }
    }
}

// ------------------------------------------------- layer 3: width-1 projection
// out[i] = dot(agg[i]/deg[i], w3l) + dot(h[i], w3r) + b3   (one wave per node)
__global__ __launch_bounds__(256) void gnn_layer3_kernel(
    const float* __restrict__ agg, const float* __restrict__ deg,
    const float* __restrict__ h,
    const float* __restrict__ w3l, const float* __restrict__ w3r,
    const float* __restrict__ b3, float* __restrict__ out, int nrows)
{
    int wid  = (blockIdx.x * blockDim.x + threadIdx.x) >> 5;
    int lane = threadIdx.x & 31;
    if (wid >= nrows) return;
    size_t base = (size_t)wid * F_DIM + (lane << 2);
    float4 am = *(const float4*)(agg + base);
    float4 ah = *(const float4*)(h + base);
    float4 wl = *(const float4*)(w3l + (lane << 2));
    float4 wr = *(const float4*)(w3r + (lane << 2));
    float inv = 1.0f / fmaxf(deg[wid], 1.0f);
    float p = (am.x * wl.x + am.y * wl.y + am.z * wl.z + am.w * wl.w) * inv
            +  ah.x * wr.x + ah.y * wr.y + ah.z * wr.z + ah.w * wr.w;
#pragma unroll
    for (int off = 16; off > 0; off >>= 1)
        p += __shfl_down(p, off, 32);
    if (lane == 0) out[wid] = p + b3[0];
}

// ---------------------------------------------------------------- launcher
extern "C" void kernel_launch(void* const* d_in, const int* in_sizes, int n_in,
                              void* d_out, int out_size, void* d_ws, size_t ws_size,
                              hipStream_t stream) {
    const float* x   = (const float*)d_in[0];
    const int*   ei  = (const int*)d_in[1];
    const float* W1l = (const float*)d_in[2];
    const float* W1r = (const float*)d_in[3];
    const float* b1  = (const float*)d_in[4];
    const float* W2l = (const float*)d_in[5];
    const float* W2r = (const float*)d_in[6];
    const float* b2  = (const float*)d_in[7];
    const float* W3l = (const float*)d_in[8];
    const float* W3r = (const float*)d_in[9];
    const float* b3  = (const float*)d_in[10];
    float* out = (float*)d_out;

    const int* src = ei;              // edge_index[0]
    const int* dst = ei + E_EDGES;    // edge_index[1]

    // workspace: deg | agg | h
    float* deg = (float*)d_ws;
    float* agg = deg + ((N_NODES + 255) & ~255);
    float* h   = agg + (size_t)N_NODES * F_DIM;

    const int nfeat = N_NODES * F_DIM;                    // 12.8M
    dim3 blk(256);
    dim3 gZeroDeg((N_NODES + 255) / 256);
    dim3 gZeroAgg((nfeat + 255) / 256);
    dim3 gDeg((E_EDGES + 255) / 256);
    dim3 gScat((E_EDGES * 32 + 255) / 256);               // 200000 blocks
    int nwaves = N_NODES / 16;                            // 6250
    dim3 gGemm((nwaves * 32 + 255) / 256);                // 782 blocks
    dim3 gL3((N_NODES * 32 + 255) / 256);                 // 12500 blocks

    // degree (shared by all layers) + layer-1 aggregation
    gnn_zero_kernel<<<gZeroDeg, blk, 0, stream>>>(deg, N_NODES);
    gnn_zero_kernel<<<gZeroAgg, blk, 0, stream>>>(agg, nfeat);
    gnn_degree_kernel<<<gDeg, blk, 0, stream>>>(dst, deg, E_EDGES);
    gnn_scatter_kernel<<<gScat, blk, 0, stream>>>(x, src, dst, agg, E_EDGES);
    gnn_sage_gemm_kernel<<<gGemm, blk, 0, stream>>>(agg, deg, x, W1l, W1r, b1,
                                                    h, 1, N_NODES);
    // layer 2 (in-place h -> h; each wave touches only its own 16 rows)
    gnn_zero_kernel<<<gZeroAgg, blk, 0, stream>>>(agg, nfeat);
    gnn_scatter_kernel<<<gScat, blk, 0, stream>>>(h, src, dst, agg, E_EDGES);
    gnn_sage_gemm_kernel<<<gGemm, blk, 0, stream>>>(agg, deg, h, W2l, W2r, b2,
                                                    h, 1, N_NODES);
    // layer 3 (project to width 1; mean over axis 1 is identity)
    gnn_zero_kernel<<<gZeroAgg, blk, 0, stream>>>(agg, nfeat);
    gnn_scatter_kernel<<<gScat, blk, 0, stream>>>(h, src, dst, agg, E_EDGES);
    gnn_layer3_kernel<<<gL3, blk, 0, stream>>>(agg, deg, h, W3l, W3r, b3,
                                               out, N_NODES);
}